// GAT_49495203119724
// MI455X (gfx1250) — compile-verified
//
#include <hip/hip_runtime.h>
#include <hip/hip_bf16.h>

// ---------------------------------------------------------------------------
// GAT on MI455X (gfx1250): dense math via V_WMMA_F32_16X16X4_F32,
// LDS staging via GLOBAL_LOAD_ASYNC_TO_LDS_B128 (ASYNCcnt) when available.
// ---------------------------------------------------------------------------

typedef __attribute__((ext_vector_type(2))) float v2f;
typedef __attribute__((ext_vector_type(8))) float v8f;

#if defined(__HIP_DEVICE_COMPILE__) &&                                   \
    __has_builtin(__builtin_amdgcn_global_load_async_to_lds_b128) &&     \
    __has_builtin(__builtin_amdgcn_s_wait_asynccnt)
#define HAVE_ASYNC 1
#else
#define HAVE_ASYNC 0
#endif

#if HAVE_ASYNC
// Builtin signature (from hipcc diagnostic): pointers to 16-byte int vectors,
// global (AS1) source and LDS (AS3) destination.
typedef int b128_t __attribute__((vector_size(16)));
typedef __attribute__((address_space(1))) b128_t* gas_b128_ptr;
typedef __attribute__((address_space(3))) b128_t* lds_b128_ptr;
#endif

__device__ __forceinline__ void async_b128(const void* g, void* l) {
#if HAVE_ASYNC
  // (global src, lds dst, imm offset, imm cpol)
  __builtin_amdgcn_global_load_async_to_lds_b128(
      (gas_b128_ptr)(void*)g, (lds_b128_ptr)l, 0, 0);
#else
  (void)g; (void)l;
#endif
}
__device__ __forceinline__ void async_wait0() {
#if HAVE_ASYNC
  __builtin_amdgcn_s_wait_asynccnt(0);
#endif
}
__device__ __forceinline__ void async_wait4() {
#if HAVE_ASYNC
  __builtin_amdgcn_s_wait_asynccnt(4);
#endif
}

__device__ __forceinline__ v8f wmma_f32_k4(v2f a, v2f b, v8f c) {
  // 8 args: (neg_a, A, neg_b, B, c_mod, C, reuse_a, reuse_b)
  return __builtin_amdgcn_wmma_f32_16x16x4_f32(
      false, a, false, b, (short)0, c, false, false);
}

static constexpr int NN = 14;       // nodes per graph
static constexpr int CC = 128;      // channels
static constexpr int EE = 64;       // random edges
static constexpr int ET = EE + NN;  // + self loops = 78
static constexpr int XP = 132;      // padded LDS row stride (16B-aligned rows)
static constexpr int AP = 20;       // padded stride for dense alpha matrix
static constexpr int KK = NN * CC;  // 1792 readout K
static constexpr float NEG_SLOPE = 0.2f;

// ---------------------------------------------------------------------------
// One fused GAT layer: block = 1 graph, 128 threads (4 waves, wave32).
// ---------------------------------------------------------------------------
template <int RELU, int WRITE_ATTN>
__global__ __launch_bounds__(128) void gat_layer_kernel(
    const float* __restrict__ xin,    // [B,14,128]
    const int* __restrict__ edges,    // [B,64,2] (src,dst)
    const float* __restrict__ W,      // [128,128]
    const float* __restrict__ a_s,    // [128]
    const float* __restrict__ a_d,    // [128]
    const float* __restrict__ bias,   // [128]
    float* __restrict__ hout,         // [B,14,128] (may alias xin; safe)
    float* __restrict__ attn)         // [B,14,14] (layer 2 only)
{
  __shared__ alignas(16) float sX[16 * XP];
  __shared__ alignas(16) float sH[16 * XP];
  __shared__ float sAgg[16 * AP];
  __shared__ float sAS[16], sAD[16];
  __shared__ float sLogit[ET], sAlpha[ET];
  __shared__ int sSrc[ET], sDst[ET];
  __shared__ float sM[16], sZ[16];

  const int b = blockIdx.x;
  const int t = threadIdx.x;
  const int lane = t & 31;
  const int w = t >> 5;

  // Hint: W (64KB) is shared by every block; keep it hot in cache.
  __builtin_prefetch(&W[t * CC], 0, 1);

  // ---- stage x tile (rows 14,15 zero-padded) ----
  const float* xg = xin + (size_t)b * NN * CC;
#if HAVE_ASYNC
  for (int i = t; i < NN * (CC / 4); i += 128) {  // 448 b128 copies
    const int r = i >> 5, c4 = (i & 31) * 4;
    async_b128(xg + r * CC + c4, &sX[r * XP + c4]);
  }
  for (int i = t; i < 2 * CC; i += 128) {  // zero pad rows 14,15
    const int r = NN + (i >> 7), c = i & 127;
    sX[r * XP + c] = 0.0f;
  }
  for (int i = t; i < 16 * AP; i += 128) sAgg[i] = 0.0f;
  async_wait0();
#else
  for (int i = t; i < 16 * (CC / 4); i += 128) {
    const int r = i >> 5, c4 = (i & 31) * 4;
    const float4 v = (r < NN) ? *(const float4*)&xg[r * CC + c4]
                              : make_float4(0.f, 0.f, 0.f, 0.f);
    *(float4*)&sX[r * XP + c4] = v;
  }
  for (int i = t; i < 16 * AP; i += 128) sAgg[i] = 0.0f;
#endif
  __syncthreads();

  // ---- GEMM: hl = x @ W via f32 WMMA (M=16, N=128, K=128) ----
  {
    const int m = lane & 15;
    const int kh = (lane >> 4) * 2;  // K-sub-offset per lane half
    const int n0 = w * 32 + (lane & 15);
    const int n1 = n0 + 16;
    v8f acc0 = {}, acc1 = {};
#pragma unroll 4
    for (int k = 0; k < CC; k += 4) {
      v2f a;
      a.x = sX[m * XP + k + kh];
      a.y = sX[m * XP + k + kh + 1];
      v2f b0, b1;
      b0.x = W[(k + kh) * CC + n0];
      b0.y = W[(k + kh + 1) * CC + n0];
      b1.x = W[(k + kh) * CC + n1];
      b1.y = W[(k + kh + 1) * CC + n1];
      acc0 = wmma_f32_k4(a, b0, acc0);
      acc1 = wmma_f32_k4(a, b1, acc1);
    }
    const int rbase = (lane >> 4) * 8;  // C/D layout: vgpr i -> row i (+8)
#pragma unroll
    for (int i = 0; i < 8; ++i) {
      sH[(rbase + i) * XP + n0] = acc0[i];
      sH[(rbase + i) * XP + n1] = acc1[i];
    }
  }
  __syncthreads();

  // ---- per-node score dots: as_n = h.a_s, ad_n = h.a_d ----
  for (int n = w; n < NN; n += 4) {
    float ps = 0.0f, pd = 0.0f;
    for (int c = lane; c < CC; c += 32) {
      const float hv = sH[n * XP + c];
      ps += hv * a_s[c];
      pd += hv * a_d[c];
    }
#pragma unroll
    for (int off = 16; off; off >>= 1) {
      ps += __shfl_xor(ps, off, 32);
      pd += __shfl_xor(pd, off, 32);
    }
    if (lane == 0) { sAS[n] = ps; sAD[n] = pd; }
  }
  __syncthreads();

  // ---- edge logits (leaky relu) ----
  if (t < ET) {
    int s, d;
    if (t < EE) {
      s = edges[(size_t)b * EE * 2 + t * 2 + 0];
      d = edges[(size_t)b * EE * 2 + t * 2 + 1];
    } else {
      s = t - EE; d = t - EE;  // self loops
    }
    sSrc[t] = s; sDst[t] = d;
    const float l = sAS[s] + sAD[d];
    sLogit[t] = (l > 0.0f) ? l : NEG_SLOPE * l;
  }
  __syncthreads();

  // ---- segment softmax grouped by dst ----
  if (t < NN) {
    float m = -3.4e38f;
    for (int e = 0; e < ET; ++e)
      if (sDst[e] == t) m = fmaxf(m, sLogit[e]);
    float z = 0.0f;
    for (int e = 0; e < ET; ++e)
      if (sDst[e] == t) z += __expf(sLogit[e] - m);
    sM[t] = m; sZ[t] = z;
  }
  __syncthreads();

  if (t < ET) {
    const int s = sSrc[t], d = sDst[t];
    const float alpha = __expf(sLogit[t] - sM[d]) / sZ[d];
    sAlpha[t] = alpha;
    atomicAdd(&sAgg[d * AP + s], alpha);  // ds_add_f32; sums duplicate edges
  }
  __syncthreads();

  if (WRITE_ATTN) {
    for (int i = t; i < NN * NN; i += 128) {
      const int s = i / NN, d = i % NN;
      float v = 0.0f;
      for (int e = 0; e < ET; ++e)
        if (sSrc[e] == s && sDst[e] == d) v = sAlpha[e];
      attn[(size_t)b * NN * NN + i] = v;
    }
  }

  // ---- aggregate: out = alpha_dense @ hl (WMMA, K=16) + bias (+ReLU) ----
  {
    const int m = lane & 15;
    const int kh = (lane >> 4) * 2;
    const int n0 = w * 32 + (lane & 15);
    const int n1 = n0 + 16;
    v8f acc0 = {}, acc1 = {};
#pragma unroll
    for (int k = 0; k < 16; k += 4) {
      v2f a;
      a.x = sAgg[m * AP + k + kh];
      a.y = sAgg[m * AP + k + kh + 1];
      v2f b0, b1;
      b0.x = sH[(k + kh) * XP + n0];
      b0.y = sH[(k + kh + 1) * XP + n0];
      b1.x = sH[(k + kh) * XP + n1];
      b1.y = sH[(k + kh + 1) * XP + n1];
      acc0 = wmma_f32_k4(a, b0, acc0);
      acc1 = wmma_f32_k4(a, b1, acc1);
    }
    const int rbase = (lane >> 4) * 8;
    float* og = hout + (size_t)b * NN * CC;
    const float bb0 = bias[n0], bb1 = bias[n1];
#pragma unroll
    for (int i = 0; i < 8; ++i) {
      const int row = rbase + i;
      if (row < NN) {
        float v0 = acc0[i] + bb0;
        float v1 = acc1[i] + bb1;
        if (RELU) { v0 = fmaxf(v0, 0.0f); v1 = fmaxf(v1, 0.0f); }
        og[row * CC + n0] = v0;
        og[row * CC + n1] = v1;
      }
    }
  }
}

// ---------------------------------------------------------------------------
// Readout: x = h2.reshape(B,1792) @ Wl[1792,64] + bl; pred = sigmoid(x@Wp+bp).
// Block = 16 graphs (M-tile), 4 waves = 4 n-tiles of 16.
// K chunked (14 x 128) through double-buffered LDS with async prefetch.
// ---------------------------------------------------------------------------
__global__ __launch_bounds__(128) void readout_kernel(
    const float* __restrict__ h2,   // [B,1792]
    const float* __restrict__ Wl,   // [1792,64]
    const float* __restrict__ bl,   // [64]
    const float* __restrict__ Wp,   // [64]
    const float* __restrict__ bp,   // [1]
    float* __restrict__ pred)       // [B]
{
  __shared__ alignas(16) float sA[2][16 * XP];  // double-buffered K-chunks
  __shared__ float sXo[16 * 68];                // x tile for final head
  const int b0 = blockIdx.x * 16;
  const int t = threadIdx.x;
  const int lane = t & 31;
  const int w = t >> 5;

  const int m = lane & 15;
  const int kh = (lane >> 4) * 2;
  const int n = w * 16 + (lane & 15);

  // stage chunk kc into buffer buf (4 x b128 per thread)
  auto stage = [&](int kc, int buf) {
#pragma unroll
    for (int j = 0; j < 4; ++j) {
      const int i = t + j * 128;
      const int r = i >> 5, c4 = (i & 31) * 4;
#if HAVE_ASYNC
      async_b128(h2 + (size_t)(b0 + r) * KK + kc * 128 + c4,
                 &sA[buf][r * XP + c4]);
#else
      *(float4*)&sA[buf][r * XP + c4] =
          *(const float4*)&h2[(size_t)(b0 + r) * KK + kc * 128 + c4];
#endif
    }
  };

  stage(0, 0);
  v8f acc = {};
  for (int kc = 0; kc < NN; ++kc) {
    const int buf = kc & 1;
    const bool more = (kc + 1) < NN;
    if (more) stage(kc + 1, buf ^ 1);   // prefetch next chunk
#if HAVE_ASYNC
    if (more) async_wait4(); else async_wait0();  // current chunk landed
#endif
    __syncthreads();
    const int kg = kc * 128;
    const float* __restrict__ sAb = sA[buf];
#pragma unroll 4
    for (int k = 0; k < 128; k += 4) {
      v2f a;
      a.x = sAb[m * XP + k + kh];
      a.y = sAb[m * XP + k + kh + 1];
      v2f bb;
      bb.x = Wl[(kg + k + kh) * 64 + n];
      bb.y = Wl[(kg + k + kh + 1) * 64 + n];
      acc = wmma_f32_k4(a, bb, acc);
    }
    __syncthreads();  // done reading before this buffer is refilled
  }

  const int rbase = (lane >> 4) * 8;
  const float blv = bl[n];
#pragma unroll
  for (int i = 0; i < 8; ++i) sXo[(rbase + i) * 68 + n] = acc[i] + blv;
  __syncthreads();

  if (t < 16) {
    float p = bp[0];
    for (int c = 0; c < 64; ++c) p += sXo[t * 68 + c] * Wp[c];
    pred[b0 + t] = 1.0f / (1.0f + __expf(-p));
  }
}

// ---------------------------------------------------------------------------
extern "C" void kernel_launch(void* const* d_in, const int* in_sizes, int n_in,
                              void* d_out, int out_size, void* d_ws,
                              size_t ws_size, hipStream_t stream) {
  const float* feature = (const float*)d_in[0];
  const int* edges     = (const int*)d_in[1];
  const float* W1  = (const float*)d_in[2];
  const float* as1 = (const float*)d_in[3];
  const float* ad1 = (const float*)d_in[4];
  const float* b1  = (const float*)d_in[5];
  const float* W2  = (const float*)d_in[6];
  const float* as2 = (const float*)d_in[7];
  const float* ad2 = (const float*)d_in[8];
  const float* b2  = (const float*)d_in[9];
  const float* Wl  = (const float*)d_in[10];
  const float* bl  = (const float*)d_in[11];
  const float* Wp  = (const float*)d_in[12];
  const float* bp  = (const float*)d_in[13];

  const int B = in_sizes[0] / (NN * CC);  // 32768

  float* pred = (float*)d_out;            // [B,1]
  float* attn = (float*)d_out + B;        // [B,14,14]
  float* h    = (float*)d_ws;             // [B,14,128] reused in place

  dim3 blk(128);
  // layer 1: ReLU, no attn output
  gat_layer_kernel<1, 0><<<B, blk, 0, stream>>>(feature, edges, W1, as1, ad1,
                                                b1, h, nullptr);
  // layer 2: no ReLU, writes attn; in-place on h (block-local staging)
  gat_layer_kernel<0, 1><<<B, blk, 0, stream>>>(h, edges, W2, as2, ad2, b2, h,
                                                attn);
  // readout + prediction head
  readout_kernel<<<B / 16, blk, 0, stream>>>(h, Wl, bl, Wp, bp, pred);
}